// FeatureUpsampler_81870666596705
// MI455X (gfx1250) — compile-verified
//
#include <hip/hip_runtime.h>

typedef __attribute__((ext_vector_type(2))) float v2f;
typedef __attribute__((ext_vector_type(4))) float v4f;
typedef __attribute__((ext_vector_type(8))) float v8f;

#define B_   4
#define N_   16384
#define M_   2048
#define D1_  128
#define D2_  256
#define K1_  384       // D1 + D2
#define C1_  256
#define C2_  256
#define J_   (B_ * N_) // 65536 fused (b,n) columns; BN stats reduce over all of them
#define BN_EPS 1e-5f

__device__ __forceinline__ v8f wmma4(v2f a, v2f b, v8f c) {
    return __builtin_amdgcn_wmma_f32_16x16x4_f32(
        /*neg_a=*/false, a, /*neg_b=*/false, b,
        /*c_mod=*/(short)0, c, /*reuse_a=*/false, /*reuse_b=*/false);
}

// ---------------------------------------------------------------------------
// 1) three_nn: per dense point, keep 3 smallest squared distances vs 2048
//    sparse points; sparse coords streamed through LDS in 256-point tiles.
//    All 256 threads of a block share one batch (16384 % 256 == 0).
// ---------------------------------------------------------------------------
__global__ __launch_bounds__(256) void three_nn_kernel(
    const float* __restrict__ dense_coords,   // [B,N,3]
    const float* __restrict__ sparse_coords,  // [B,M,3]
    int*   __restrict__ idx3,                 // [B*N,3]
    float* __restrict__ w3)                   // [B*N,3]
{
    __shared__ float sc[256 * 3];
    const int t = threadIdx.x;
    const int j = blockIdx.x * 256 + t;
    const int b = j >> 14;

    const float px = dense_coords[j * 3 + 0];
    const float py = dense_coords[j * 3 + 1];
    const float pz = dense_coords[j * 3 + 2];

    float b0 = 3.4e38f, b1 = 3.4e38f, b2 = 3.4e38f;
    int   i0 = 0, i1 = 0, i2 = 0;

    for (int m0 = 0; m0 < M_; m0 += 256) {
        __syncthreads();
        const float* src = sparse_coords + (size_t)(b * M_ + m0) * 3;
        for (int i = t; i < 768; i += 256) sc[i] = src[i];
        __syncthreads();
        #pragma unroll 4
        for (int mm = 0; mm < 256; ++mm) {
            const float dx = px - sc[mm * 3 + 0];
            const float dy = py - sc[mm * 3 + 1];
            const float dz = pz - sc[mm * 3 + 2];
            const float d  = dx * dx + dy * dy + dz * dz;
            const int   m  = m0 + mm;
            if (d < b2) {
                if (d < b1) {
                    if (d < b0) { b2 = b1; i2 = i1; b1 = b0; i1 = i0; b0 = d; i0 = m; }
                    else        { b2 = b1; i2 = i1; b1 = d;  i1 = m; }
                } else          { b2 = d;  i2 = m; }
            }
        }
    }
    const float d0 = sqrtf(fmaxf(b0, 0.f));
    const float d1 = sqrtf(fmaxf(b1, 0.f));
    const float d2 = sqrtf(fmaxf(b2, 0.f));
    const float r0 = 1.f / (d0 + 1e-8f);
    const float r1 = 1.f / (d1 + 1e-8f);
    const float r2 = 1.f / (d2 + 1e-8f);
    const float rs = 1.f / (r0 + r1 + r2);
    idx3[j * 3 + 0] = i0; idx3[j * 3 + 1] = i1; idx3[j * 3 + 2] = i2;
    w3[j * 3 + 0] = r0 * rs; w3[j * 3 + 1] = r1 * rs; w3[j * 3 + 2] = r2 * rs;
}

// ---------------------------------------------------------------------------
// 2) Pack W[O][K] row-major -> Wp[K/4][O][4] (WMMA A-fragment order: one
//    contiguous b64 per lane per k-step). Tiny, runs once per layer.
// ---------------------------------------------------------------------------
__global__ __launch_bounds__(256) void pack_w_kernel(
    const float* __restrict__ W, float* __restrict__ Wp, int O, int K)
{
    const int t = blockIdx.x * 256 + threadIdx.x;   // over (K/4)*O
    const int K4 = K >> 2;
    if (t >= K4 * O) return;
    const int kk4 = t / O, o = t - kk4 * O;
    const float* src = W + (size_t)o * K + 4 * kk4;
    v4f v; v[0] = src[0]; v[1] = src[1]; v[2] = src[2]; v[3] = src[3];
    *(v4f*)(Wp + (size_t)t * 4) = v;
}

// ---------------------------------------------------------------------------
// 3) interpolation + concat, written directly in B-fragment order:
//    Xp[k/4][j][4] with Xp[kk4][j][c] = combined[4*kk4+c][j].
//    One thread per (k4, j): computes 4 channels, stores one b128 (coalesced).
// ---------------------------------------------------------------------------
__global__ __launch_bounds__(256) void interp_concat_kernel(
    const float* __restrict__ dense_features,   // [B,D1,N]
    const float* __restrict__ sparse_features,  // [B,D2,M]
    const int*   __restrict__ idx3,
    const float* __restrict__ w3,
    float*       __restrict__ Xp)               // [K1/4][J][4]
{
    const long t  = (long)blockIdx.x * 256 + threadIdx.x;  // over (K1/4)*J
    const int  k4 = (int)(t >> 16);                        // J == 65536
    const int  j  = (int)(t & 65535);
    const int  b  = j >> 14, n = j & 16383;
    v4f v;
    if (k4 < D1_ / 4) {
        const float* src = dense_features + ((size_t)b * D1_ + 4 * k4) * N_ + n;
        v[0] = src[0];
        v[1] = src[(size_t)N_];
        v[2] = src[(size_t)2 * N_];
        v[3] = src[(size_t)3 * N_];
    } else {
        const int d = 4 * k4 - D1_;
        const float* f = sparse_features + ((size_t)b * D2_ + d) * M_;
        const int ia = idx3[j * 3 + 0], ib = idx3[j * 3 + 1], ic = idx3[j * 3 + 2];
        const float wa = w3[j * 3 + 0], wb = w3[j * 3 + 1], wc = w3[j * 3 + 2];
        #pragma unroll
        for (int c = 0; c < 4; ++c) {
            const float* fc = f + (size_t)c * M_;
            v[c] = fc[ia] * wa + fc[ib] * wb + fc[ic] * wc;
        }
    }
    *(v4f*)(Xp + (size_t)t * 4) = v;   // (k4*J + j)*4 == t*4
}

// ---------------------------------------------------------------------------
// 4) GEMM  Y[O][J] = W[O][K] @ X[K][J]  via V_WMMA_F32_16X16X4_F32.
//    Block: 8 waves, one 16-row stripe, 512 columns. Each wave owns 4
//    independent 16x16 tiles (4 accumulators sharing one A fragment).
//    The packed A stripe (K*64 bytes) is staged in LDS with
//    GLOBAL_LOAD_ASYNC_TO_LDS_B128 (ASYNCcnt path), then read back as
//    conflict-free ds_load_b64 fragments. k-loop is 1-deep pipelined.
// ---------------------------------------------------------------------------
template <int K>
__global__ __launch_bounds__(256) void gemm_wmma_kernel(
    const float* __restrict__ Wp,  // [K/4][O][4] packed A fragments
    const float* __restrict__ Xp,  // [K/4][J][4] packed B fragments
    float*       __restrict__ Y,   // [O][J] row-major
    int O)
{
    constexpr int K4 = K / 4;
    __shared__ float sA[K4 * 16 * 4];   // A stripe, fragment order

    const int tid  = threadIdx.x;
    const int wave = tid >> 5;
    const int lane = tid & 31;
    const int half = lane >> 4;        // selects K-pair {0,1} vs {2,3}
    const int l    = lane & 15;        // row (A) / column (B,C,D) in tile
    const int row0 = blockIdx.y * 16;
    const int col0 = blockIdx.x * 512 + wave * 64;

    // ---- async DMA: stage packed A stripe (rows row0..row0+15) into LDS
    const unsigned lds_base = (unsigned)(unsigned long long)(void*)sA; // low 32b of
                                                                       // LDS aperture addr == LDS offset
    for (int e = tid; e < K4 * 16; e += 256) {
        const int kk4 = e >> 4, r = e & 15;
        const float* g = Wp + ((size_t)kk4 * O + row0 + r) * 4;
        asm volatile("global_load_async_to_lds_b128 %0, %1, off"
                     :: "v"(lds_base + (unsigned)e * 16u),
                        "v"((unsigned long long)g)
                     : "memory");
    }
    asm volatile("s_wait_asynccnt 0x0" ::: "memory");
    __syncthreads();

    auto ldA = [&](int kk4) -> v2f {
        return *(const v2f*)(&sA[(kk4 * 16 + l) * 4 + 2 * half]);
    };
    auto ldB = [&](int kk4, int t) -> v2f {
        return *(const v2f*)(Xp + (((size_t)kk4 * J_) + col0 + 16 * t + l) * 4 + 2 * half);
    };

    v8f acc[4] = {{}, {}, {}, {}};
    v2f a = ldA(0);
    v2f b[4] = {ldB(0, 0), ldB(0, 1), ldB(0, 2), ldB(0, 3)};

    #pragma unroll 2
    for (int kk4 = 0; kk4 < K4; ++kk4) {
        const int nx = (kk4 + 1 < K4) ? kk4 + 1 : kk4;
        v2f an = ldA(nx);
        v2f bn[4] = {ldB(nx, 0), ldB(nx, 1), ldB(nx, 2), ldB(nx, 3)};
        #pragma unroll
        for (int t = 0; t < 4; ++t) acc[t] = wmma4(a, b[t], acc[t]);
        a = an;
        #pragma unroll
        for (int t = 0; t < 4; ++t) b[t] = bn[t];
    }

    #pragma unroll
    for (int t = 0; t < 4; ++t) {
        float* yo = Y + (size_t)(row0 + 8 * half) * J_ + col0 + 16 * t + l;
        #pragma unroll
        for (int r = 0; r < 8; ++r) yo[(size_t)r * J_] = acc[t][r];  // VGPR r -> row r+8*half
    }
}

// ---------------------------------------------------------------------------
// 5) BatchNorm stats: one block per output channel, reduce sum/sumsq over J.
// ---------------------------------------------------------------------------
__global__ __launch_bounds__(256) void bn_stats_kernel(
    const float* __restrict__ Y, float* __restrict__ sum, float* __restrict__ sumsq)
{
    __shared__ float s1[256], s2[256];
    const int o = blockIdx.x;
    const float* row = Y + (size_t)o * J_;
    float a = 0.f, q = 0.f;
    for (int i = threadIdx.x; i < J_; i += 256) { const float v = row[i]; a += v; q += v * v; }
    s1[threadIdx.x] = a; s2[threadIdx.x] = q;
    __syncthreads();
    for (int st = 128; st > 0; st >>= 1) {
        if (threadIdx.x < st) {
            s1[threadIdx.x] += s1[threadIdx.x + st];
            s2[threadIdx.x] += s2[threadIdx.x + st];
        }
        __syncthreads();
    }
    if (threadIdx.x == 0) { sum[o] = s1[0]; sumsq[o] = s2[0]; }
}

// ---------------------------------------------------------------------------
// 6a) normalize + ReLU, output packed in B-fragment order for the next GEMM:
//     Xp[o/4][j][4]. One thread per (o4, j), b128 store.
// ---------------------------------------------------------------------------
__global__ __launch_bounds__(256) void bn_relu_pack_kernel(
    const float* __restrict__ Y,
    const float* __restrict__ sum, const float* __restrict__ sumsq,
    const float* __restrict__ gamma, const float* __restrict__ beta,
    float* __restrict__ Xp)
{
    const long t  = (long)blockIdx.x * 256 + threadIdx.x;  // over (C1/4)*J
    const int  o4 = (int)(t >> 16);
    const int  j  = (int)(t & 65535);
    const float inv = 1.0f / (float)J_;
    v4f v;
    #pragma unroll
    for (int c = 0; c < 4; ++c) {
        const int o = 4 * o4 + c;
        const float m   = sum[o] * inv;
        const float var = fmaxf(sumsq[o] * inv - m * m, 0.f);
        const float sc  = gamma[o] * rsqrtf(var + BN_EPS);
        v[c] = fmaxf((Y[(size_t)o * J_ + j] - m) * sc + beta[o], 0.f);
    }
    *(v4f*)(Xp + (size_t)t * 4) = v;
}

// ---------------------------------------------------------------------------
// 6b) final normalize + ReLU with layout remap [O][b*N+n] -> [B,O,N].
// ---------------------------------------------------------------------------
__global__ __launch_bounds__(256) void bn_relu_out_kernel(
    const float* __restrict__ Y,
    const float* __restrict__ sum, const float* __restrict__ sumsq,
    const float* __restrict__ gamma, const float* __restrict__ beta,
    float* __restrict__ out)
{
    const long t = (long)blockIdx.x * 256 + threadIdx.x;  // over C2*J
    const int  o = (int)(t >> 16);
    const int  j = (int)(t & 65535);
    const float inv = 1.0f / (float)J_;
    const float m   = sum[o] * inv;
    const float var = fmaxf(sumsq[o] * inv - m * m, 0.f);
    const float sc  = gamma[o] * rsqrtf(var + BN_EPS);
    const float v   = fmaxf((Y[t] - m) * sc + beta[o], 0.f);
    const int b = j >> 14, n = j & 16383;
    out[((size_t)b * C2_ + o) * N_ + n] = v;
}

// ---------------------------------------------------------------------------
extern "C" void kernel_launch(void* const* d_in, const int* in_sizes, int n_in,
                              void* d_out, int out_size, void* d_ws, size_t ws_size,
                              hipStream_t stream) {
    (void)in_sizes; (void)n_in; (void)out_size; (void)ws_size;

    const float* dense_coords    = (const float*)d_in[0];
    const float* sparse_coords   = (const float*)d_in[1];
    const float* dense_features  = (const float*)d_in[2];
    const float* sparse_features = (const float*)d_in[3];
    const float* W1  = (const float*)d_in[4];
    const float* g1  = (const float*)d_in[5];
    const float* be1 = (const float*)d_in[6];
    const float* W2  = (const float*)d_in[7];
    const float* g2  = (const float*)d_in[8];
    const float* be2 = (const float*)d_in[9];
    float* out = (float*)d_out;

    char*  ws  = (char*)d_ws;
    size_t off = 0;
    auto alloc = [&](size_t bytes) -> char* {
        char* p = ws + off;
        off += (bytes + 255) & ~(size_t)255;
        return p;
    };
    int*   idx3 = (int*)  alloc((size_t)J_ * 3 * sizeof(int));
    float* w3   = (float*)alloc((size_t)J_ * 3 * sizeof(float));
    float* sum1 = (float*)alloc(C1_ * sizeof(float));
    float* sq1  = (float*)alloc(C1_ * sizeof(float));
    float* sum2 = (float*)alloc(C2_ * sizeof(float));
    float* sq2  = (float*)alloc(C2_ * sizeof(float));
    float* Wp1  = (float*)alloc((size_t)C1_ * K1_ * sizeof(float));
    float* Wp2  = (float*)alloc((size_t)C2_ * C1_ * sizeof(float));
    float* Xp1  = (float*)alloc((size_t)K1_ * J_ * sizeof(float));  // packed combined
    float* Y1   = (float*)alloc((size_t)C1_ * J_ * sizeof(float));
    float* Xp2  = (float*)alloc((size_t)C1_ * J_ * sizeof(float));  // packed hidden
    float* Y2   = Y1;  // gemm2 only reads Xp2, so Y2 can alias Y1

    pack_w_kernel<<<(C1_ * (K1_ / 4) + 255) / 256, 256, 0, stream>>>(W1, Wp1, C1_, K1_);
    pack_w_kernel<<<(C2_ * (C1_ / 4) + 255) / 256, 256, 0, stream>>>(W2, Wp2, C2_, C1_);

    three_nn_kernel<<<J_ / 256, 256, 0, stream>>>(dense_coords, sparse_coords, idx3, w3);

    interp_concat_kernel<<<(K1_ / 4) * (J_ / 256), 256, 0, stream>>>(
        dense_features, sparse_features, idx3, w3, Xp1);

    gemm_wmma_kernel<K1_><<<dim3(J_ / 512, C1_ / 16), 256, 0, stream>>>(Wp1, Xp1, Y1, C1_);
    bn_stats_kernel<<<C1_, 256, 0, stream>>>(Y1, sum1, sq1);
    bn_relu_pack_kernel<<<(C1_ / 4) * (J_ / 256), 256, 0, stream>>>(Y1, sum1, sq1, g1, be1, Xp2);

    gemm_wmma_kernel<C1_><<<dim3(J_ / 512, C2_ / 16), 256, 0, stream>>>(Wp2, Xp2, Y2, C2_);
    bn_stats_kernel<<<C2_, 256, 0, stream>>>(Y2, sum2, sq2);
    bn_relu_out_kernel<<<C2_ * (J_ / 256), 256, 0, stream>>>(Y2, sum2, sq2, g2, be2, out);
}